// AdaptivePointNet2FeaturePropagator_72705206386969
// MI455X (gfx1250) — compile-verified
//
#include <hip/hip_runtime.h>
#include <hip/hip_bf16.h>

// ---------------------------------------------------------------------------
// AdaptivePointNet2FeaturePropagator for MI455X (gfx1250, wave32, WMMA)
//
//   1. frame_ranges : binary-search per-frame query ranges (sorted indices)
//   2. knn          : per-point top-3 within its frame (~512 scans/pt)
//   3. interp       : inverse-distance weighted gather of features_prev
//   4. gemm1 (WMMA f32 16x16x4, 64x16 tile/wave, SW-pipelined):
//        Y1 = W1 @ [interp; features] + b1
//   5. bn_stats(Y1) -> alpha1/beta1
//   6. gemm2 (WMMA, BN1+ReLU fused into B loads): Y2raw -> d_out
//   7. bn_stats(d_out) -> alpha2/beta2 ; bn_apply in-place on d_out
// ---------------------------------------------------------------------------

#define FRAMES   8
#define BIG_D    1e10f
#define BN_EPS   1e-5f
#define DIST_EPS 1e-8f

typedef float v2f __attribute__((ext_vector_type(2)));
typedef float v8f __attribute__((ext_vector_type(8)));

// -------------------------------------------------------------------- ranges
__global__ void frame_ranges_kernel(const int* __restrict__ qf, int M,
                                    int* __restrict__ ranges) {
    int f = threadIdx.x;
    if (f >= FRAMES) return;
    int lo = 0, hi = M;
    while (lo < hi) { int mid = (lo + hi) >> 1; if (qf[mid] < f) lo = mid + 1; else hi = mid; }
    ranges[f] = lo;
    lo = 0; hi = M;
    while (lo < hi) { int mid = (lo + hi) >> 1; if (qf[mid] < f + 1) lo = mid + 1; else hi = mid; }
    ranges[FRAMES + f] = lo;
}

// ----------------------------------------------------------------------- knn
__global__ void __launch_bounds__(256)
knn_kernel(const float* __restrict__ xyz, const float* __restrict__ xyzp,
           const int* __restrict__ pf, const int* __restrict__ ranges,
           int N, int M, int* __restrict__ idx4, float* __restrict__ w4) {
    int n = blockIdx.x * 256 + threadIdx.x;
    if (n >= N) return;
    float px = xyz[3*n+0], py = xyz[3*n+1], pz = xyz[3*n+2];
    int f = pf[n];
    int s = ranges[f], e = ranges[FRAMES + f];
    float d0 = BIG_D, d1 = BIG_D, d2 = BIG_D;
    int   i0 = 0, i1 = (M > 1) ? 1 : 0, i2 = (M > 2) ? 2 : 0;
    for (int j = s; j < e; ++j) {          // lanes share frame (sorted) -> broadcast loads
        float dx = px - xyzp[3*j+0];
        float dy = py - xyzp[3*j+1];
        float dz = pz - xyzp[3*j+2];
        float d  = dx*dx + dy*dy + dz*dz;
        if (d < d0)      { d2=d1; i2=i1; d1=d0; i1=i0; d0=d; i0=j; }
        else if (d < d1) { d2=d1; i2=i1; d1=d;  i1=j; }
        else if (d < d2) { d2=d;  i2=j; }
    }
    // recompute true distances from gathered coords (matches reference; also
    // handles degenerate frames where defaults 0,1,2 were kept)
    int ii[3] = { i0, i1, i2 };
    float inv[3];
    float invsum = 0.f;
    #pragma unroll
    for (int k = 0; k < 3; ++k) {
        float dx = px - xyzp[3*ii[k]+0];
        float dy = py - xyzp[3*ii[k]+1];
        float dz = pz - xyzp[3*ii[k]+2];
        float dist = sqrtf(dx*dx + dy*dy + dz*dz);
        inv[k] = 1.f / (dist + DIST_EPS);
        invsum += inv[k];
    }
    float rs = 1.f / invsum;
    idx4[4*n+0] = i0; idx4[4*n+1] = i1; idx4[4*n+2] = i2; idx4[4*n+3] = 0;
    w4[4*n+0] = inv[0]*rs; w4[4*n+1] = inv[1]*rs; w4[4*n+2] = inv[2]*rs; w4[4*n+3] = 0.f;
}

// -------------------------------------------------------------------- interp
__global__ void __launch_bounds__(256)
interp_kernel(const float* __restrict__ fp /*Cp x M*/,
              const int* __restrict__ idx4, const float* __restrict__ w4,
              int N, int M, int Cp, float* __restrict__ X0 /*Cp x N*/) {
    int n = blockIdx.x * 256 + threadIdx.x;
    if (n >= N) return;
    int   i0 = idx4[4*n+0], i1 = idx4[4*n+1], i2 = idx4[4*n+2];
    float w0 = w4[4*n+0],  w1 = w4[4*n+1],  w2 = w4[4*n+2];
    for (int c = 0; c < Cp; ++c) {
        const float* __restrict__ row = fp + (size_t)c * M;      // 16KB row, L2 resident
        X0[(size_t)c * N + n] = w0*row[i0] + w1*row[i1] + w2*row[i2];  // coalesced store
    }
}

// -------------------------------------------------- GEMM via V_WMMA f32 16x16x4
// One wave computes 4 row-tiles x 1 col-tile (64x16 f32 output).
// Software-pipelined: the k-step's loads (4x A b64, 2x B b32 [, BN coeffs])
// are issued one iteration ahead of the 4 WMMAs that consume them, so the
// wait before a WMMA covers loads launched a full iteration earlier.
struct KFrag {
    v2f   a0, a1, a2, a3;      // A 16x4 fragments for the 4 row tiles
    float b0, b1;              // B 4x16 fragment (raw)
    float al0, al1, bt0, bt1;  // BN coeffs (FUSE only)
};

template<bool FUSE>
__device__ __forceinline__ void load_frag(
        KFrag& f,
        const float* __restrict__ ap, size_t AstrideRT,
        const float* __restrict__ bp, int Nc,
        const float* __restrict__ alpha, const float* __restrict__ beta,
        int krr) {
    f.a0 = *(const v2f*)(ap);
    f.a1 = *(const v2f*)(ap + AstrideRT);
    f.a2 = *(const v2f*)(ap + 2 * AstrideRT);
    f.a3 = *(const v2f*)(ap + 3 * AstrideRT);
    f.b0 = bp[0];
    f.b1 = bp[Nc];
    if (FUSE) {
        f.al0 = alpha[krr];     f.bt0 = beta[krr];
        f.al1 = alpha[krr + 1]; f.bt1 = beta[krr + 1];
    }
}

template<bool FUSE>
__device__ __forceinline__ void compute_frag(v8f acc[4], const KFrag& f) {
    float b0 = f.b0, b1 = f.b1;
    if (FUSE) {
        b0 = fmaxf(b0 * f.al0 + f.bt0, 0.f);
        b1 = fmaxf(b1 * f.al1 + f.bt1, 0.f);
    }
    v2f b; b.x = b0; b.y = b1;
    acc[0] = __builtin_amdgcn_wmma_f32_16x16x4_f32(false, f.a0, false, b, (short)0, acc[0], false, false);
    acc[1] = __builtin_amdgcn_wmma_f32_16x16x4_f32(false, f.a1, false, b, (short)0, acc[1], false, false);
    acc[2] = __builtin_amdgcn_wmma_f32_16x16x4_f32(false, f.a2, false, b, (short)0, acc[2], false, false);
    acc[3] = __builtin_amdgcn_wmma_f32_16x16x4_f32(false, f.a3, false, b, (short)0, acc[3], false, false);
}

template<bool FUSE>
__device__ __forceinline__ void gemm_segment(
        v8f acc[4],
        const float* __restrict__ Abase0,   // A + (m0+l)*K + 2*half
        size_t AstrideRT,
        int kbeg, int kend,                 // absolute K range within A, len % 4 == 0, >= 4
        const float* __restrict__ B,        // segment B, rows [0, kend-kbeg)
        int Nc, int coloff,                 // coloff = n0 + l
        const float* __restrict__ alpha, const float* __restrict__ beta,
        int half) {
    const int seglen = kend - kbeg;
    const float* __restrict__ ap = Abase0 + kbeg;
    const float* __restrict__ bp = B + (size_t)(2 * half) * Nc + coloff;
    const int krr0 = 2 * half;

    KFrag cur, nxt;
    load_frag<FUSE>(cur, ap, AstrideRT, bp, Nc, alpha, beta, krr0);

    int kr = 0;
    for (; kr + 4 < seglen; kr += 4) {
        // issue next iteration's loads before this iteration's WMMAs
        load_frag<FUSE>(nxt, ap + (kr + 4), AstrideRT,
                        bp + (size_t)(kr + 4) * Nc, Nc,
                        alpha, beta, krr0 + kr + 4);
        __builtin_prefetch(bp + (size_t)(kr + 12) * Nc, 0, 3);  // near-cache B prefetch
        compute_frag<FUSE>(acc, cur);
        cur = nxt;
    }
    compute_frag<FUSE>(acc, cur);   // peeled last step
}

template<bool FUSE>
__global__ void __launch_bounds__(128)
gemm_wmma_kernel(const float* __restrict__ A, int K,
                 const float* __restrict__ B0, int K0,   // rows [0,K0) of B
                 const float* __restrict__ B1,           // rows [K0,K) of B
                 const float* __restrict__ bias,
                 const float* __restrict__ alpha,
                 const float* __restrict__ beta,
                 int Nc, float* __restrict__ Y) {
    const int lane = threadIdx.x & 31;
    const int wave = threadIdx.x >> 5;
    const int half = lane >> 4;               // K-split across half-waves
    const int l    = lane & 15;

    const int n0 = (blockIdx.x * 4 + wave) * 16;   // col tile (16 cols)
    const int m0 = blockIdx.y * 64;                // row group (4 tiles of 16)

    v8f acc[4] = {};

    const float* Abase0 = A + (size_t)(m0 + l) * K + 2 * half;
    const size_t AstrideRT = (size_t)16 * K;

    gemm_segment<FUSE>(acc, Abase0, AstrideRT, 0, K0, B0, Nc, n0 + l,
                       alpha, beta, half);
    if (K > K0)
        gemm_segment<FUSE>(acc, Abase0, AstrideRT, K0, K, B1, Nc, n0 + l,
                           alpha, beta, half);

    // C/D layout: VGPR r -> M = r + half*8, N = lane&15
    #pragma unroll
    for (int rt = 0; rt < 4; ++rt) {
        #pragma unroll
        for (int r = 0; r < 8; ++r) {
            int row = m0 + rt * 16 + half * 8 + r;
            Y[(size_t)row * Nc + n0 + l] = acc[rt][r] + bias[row];
        }
    }
}

// ------------------------------------------------------------------ BN stats
__global__ void __launch_bounds__(256)
bn_stats_kernel(const float* __restrict__ Y, int Nc,
                const float* __restrict__ g, const float* __restrict__ be,
                float* __restrict__ alpha, float* __restrict__ beta) {
    __shared__ float ssum[256], ssq[256];
    int row = blockIdx.x;
    const float* __restrict__ yr = Y + (size_t)row * Nc;
    float s = 0.f, q = 0.f;
    for (int i = threadIdx.x; i < Nc; i += 256) { float v = yr[i]; s += v; q += v * v; }
    ssum[threadIdx.x] = s; ssq[threadIdx.x] = q;
    __syncthreads();
    for (int off = 128; off > 0; off >>= 1) {
        if (threadIdx.x < off) {
            ssum[threadIdx.x] += ssum[threadIdx.x + off];
            ssq[threadIdx.x]  += ssq[threadIdx.x + off];
        }
        __syncthreads();
    }
    if (threadIdx.x == 0) {
        float mean = ssum[0] / (float)Nc;
        float var  = ssq[0] / (float)Nc - mean * mean;
        float rs   = rsqrtf(var + BN_EPS);
        float a    = g[row] * rs;
        alpha[row] = a;
        beta[row]  = be[row] - mean * a;
    }
}

// ------------------------------------------------------------------ BN apply
__global__ void __launch_bounds__(256)
bn_apply_kernel(float* __restrict__ Y, int Nc,
                const float* __restrict__ alpha, const float* __restrict__ beta) {
    int row = blockIdx.y;
    int col = blockIdx.x * 256 + threadIdx.x;
    if (col < Nc) {
        size_t o = (size_t)row * Nc + col;
        Y[o] = fmaxf(Y[o] * alpha[row] + beta[row], 0.f);
    }
}

// ------------------------------------------------------------------ launcher
extern "C" void kernel_launch(void* const* d_in, const int* in_sizes, int n_in,
                              void* d_out, int out_size, void* d_ws, size_t ws_size,
                              hipStream_t stream) {
    const float* xyz      = (const float*)d_in[0];
    const float* xyz_prev = (const float*)d_in[1];
    const float* features = (const float*)d_in[2];   // C  x N
    const float* feat_prev= (const float*)d_in[3];   // Cp x M
    const int*   p2f      = (const int*)d_in[4];
    const int*   q2f      = (const int*)d_in[5];
    const float* W1 = (const float*)d_in[6];
    const float* b1 = (const float*)d_in[7];
    const float* g1 = (const float*)d_in[8];
    const float* be1= (const float*)d_in[9];
    const float* W2 = (const float*)d_in[10];
    const float* b2 = (const float*)d_in[11];
    const float* g2 = (const float*)d_in[12];
    const float* be2= (const float*)d_in[13];

    const int N   = in_sizes[0] / 3;          // 16384
    const int M   = in_sizes[1] / 3;          // 4096
    const int Cp  = in_sizes[3] / M;          // 256
    const int D1  = in_sizes[7];              // 256
    const int Cin = in_sizes[6] / D1;         // 384
    const int D2  = in_sizes[11];             // 256

    // workspace carve-up (256B aligned)
    char* ws = (char*)d_ws;
    size_t off = 0;
    auto carve = [&](size_t bytes) -> char* {
        char* p = ws + off;
        off = (off + bytes + 255) & ~(size_t)255;
        return p;
    };
    int*   ranges = (int*)  carve(2 * FRAMES * sizeof(int));
    int*   idx4   = (int*)  carve((size_t)N * 4 * sizeof(int));
    float* w4     = (float*)carve((size_t)N * 4 * sizeof(float));
    float* X0     = (float*)carve((size_t)Cp * N * sizeof(float));   // interp rows
    float* Y1     = (float*)carve((size_t)D1 * N * sizeof(float));   // layer-1 raw
    float* alpha1 = (float*)carve(D1 * sizeof(float));
    float* beta1  = (float*)carve(D1 * sizeof(float));
    float* alpha2 = (float*)carve(D2 * sizeof(float));
    float* beta2  = (float*)carve(D2 * sizeof(float));
    float* out    = (float*)d_out;                                   // D2 x N
    (void)ws_size; (void)n_in; (void)out_size;

    // 1. per-frame query ranges
    frame_ranges_kernel<<<1, 32, 0, stream>>>(q2f, M, ranges);

    // 2. KNN top-3 within frame
    knn_kernel<<<(N + 255) / 256, 256, 0, stream>>>(
        xyz, xyz_prev, p2f, ranges, N, M, idx4, w4);

    // 3. weighted interpolation -> X0 (Cp x N)
    interp_kernel<<<(N + 255) / 256, 256, 0, stream>>>(
        feat_prev, idx4, w4, N, M, Cp, X0);

    // 4. GEMM1: Y1 = W1 @ [X0 ; features] + b1   (implicit concat, two segments)
    gemm_wmma_kernel<false><<<dim3(N / 64, D1 / 64), 128, 0, stream>>>(
        W1, Cin, X0, Cp, features, b1, alpha1, beta1, N, Y1);

    // 5. BN1 stats
    bn_stats_kernel<<<D1, 256, 0, stream>>>(Y1, N, g1, be1, alpha1, beta1);

    // 6. GEMM2 with BN1+ReLU fused into B loads: out_raw = W2 @ relu(bn(Y1)) + b2
    gemm_wmma_kernel<true><<<dim3(N / 64, D2 / 64), 128, 0, stream>>>(
        W2, D1, Y1, D1, Y1, b2, alpha1, beta1, N, out);

    // 7. BN2 stats + in-place apply (relu)
    bn_stats_kernel<<<D2, 256, 0, stream>>>(out, N, g2, be2, alpha2, beta2);
    bn_apply_kernel<<<dim3((N + 255) / 256, D2), 256, 0, stream>>>(
        out, N, alpha2, beta2);
}